// Latent_32160715112887
// MI455X (gfx1250) — compile-verified
//
#include <hip/hip_runtime.h>
#include <hip/hip_bf16.h>
#include <math.h>

typedef __attribute__((ext_vector_type(2))) float v2f;
typedef __attribute__((ext_vector_type(8))) float v8f;

#define N_ROWS 16384
#define M_COLS 4096
#define ZD 64
#define HW 1024            // H*W
#define ROW_TILES (N_ROWS / 16)
#define WAVES 8

__device__ __forceinline__ v8f wmma_f32(v2f a, v2f b, v8f c) {
  // D = A(16x4 f32) x B(4x16 f32) + C(16x16 f32)
  return __builtin_amdgcn_wmma_f32_16x16x4_f32(
      /*neg_a=*/false, a, /*neg_b=*/false, b,
      /*c_mod=*/(short)0, c, /*reuse_a=*/false, /*reuse_b=*/false);
}

// ---------------------------------------------------------------------------
// Pass 1: per-column max over belong * distance.
// Workgroup = 16 columns (exclusive), 8 waves stripe the 1024 row tiles.
// ---------------------------------------------------------------------------
__global__ __launch_bounds__(256) void pass1_colmax(
    const float* __restrict__ z, const float* __restrict__ e,
    const float* __restrict__ belong, float* __restrict__ batch_max) {
  __shared__ float smax[WAVES][16];

  const int lane   = threadIdx.x & 31;
  const int wave   = threadIdx.x >> 5;
  const int halo   = lane >> 4;      // 0 or 1 (which half-wave)
  const int lane16 = lane & 15;
  const int col_base = blockIdx.x * 16;
  const int col  = col_base + lane16;
  const int koff = halo * 2;         // half-lanes split each k-quad

  // B fragments (e^T tile) resident for whole kernel + ||e[col]||^2
  v2f bfrag[16];
  float en = 0.f;
#pragma unroll
  for (int kk = 0; kk < 16; ++kk) {
    int k = kk * 4 + koff;
    float b0 = e[col * ZD + k];
    float b1 = e[col * ZD + k + 1];
    bfrag[kk].x = b0; bfrag[kk].y = b1;
    en += b0 * b0 + b1 * b1;
  }
  en += __shfl_xor(en, 16, 32);      // combine k-halves -> full norm

  float colmax = 0.f;                // belong>=0, dist>=0

  for (int rt = wave; rt < ROW_TILES; rt += WAVES) {
    const int row_base = rt * 16;
    const int row = row_base + lane16;
    // zf[row][k] = z[b][k][h][w] -> base + k*HW
    const float* zrow = z + (row >> 10) * (ZD * HW) + (row & (HW - 1));

    v8f acc = {0.f, 0.f, 0.f, 0.f, 0.f, 0.f, 0.f, 0.f};
    float zn = 0.f;
#pragma unroll
    for (int kk = 0; kk < 16; ++kk) {
      int k = kk * 4 + koff;
      float a0 = zrow[k * HW];
      float a1 = zrow[(k + 1) * HW];
      v2f a; a.x = a0; a.y = a1;
      zn += a0 * a0 + a1 * a1;
      acc = wmma_f32(a, bfrag[kk], acc);
    }
    zn += __shfl_xor(zn, 16, 32);    // full ||z[row]||^2

#pragma unroll
    for (int j = 0; j < 8; ++j) {
      int rj = j + halo * 8;         // relative row held by acc[j] (C layout)
      float znj  = __shfl(zn, rj, 32);
      float d2   = znj + en - 2.f * acc[j];
      float dist = sqrtf(fmaxf(d2, 0.f));
      // streamed once -> non-temporal, keep z resident in L2
      float bl = __builtin_nontemporal_load(
          belong + (size_t)(row_base + rj) * M_COLS + col);
      colmax = fmaxf(colmax, bl * dist);
    }
  }

  colmax = fmaxf(colmax, __shfl_xor(colmax, 16, 32));
  if (lane < 16) smax[wave][lane16] = colmax;
  __syncthreads();
  if (threadIdx.x < 16) {
    float m = smax[0][threadIdx.x];
#pragma unroll
    for (int w = 1; w < WAVES; ++w) m = fmaxf(m, smax[w][threadIdx.x]);
    batch_max[col_base + threadIdx.x] = m;
  }
}

// ---------------------------------------------------------------------------
// Pass 2+3: EMA/clip of max, recompute distances via WMMA, streaming
// (stable) logsumexp per column over all N rows.
// ---------------------------------------------------------------------------
__global__ __launch_bounds__(256) void pass3_lse(
    const float* __restrict__ z, const float* __restrict__ e,
    const float* __restrict__ batch_max, const float* __restrict__ max_distance,
    const float* __restrict__ log_sigma, const int* __restrict__ distance_p,
    float* __restrict__ lse_out) {
  __shared__ float s_nm[16];
  __shared__ float smx[WAVES][16];
  __shared__ float ssm[WAVES][16];

  const int lane   = threadIdx.x & 31;
  const int wave   = threadIdx.x >> 5;
  const int halo   = lane >> 4;
  const int lane16 = lane & 15;
  const int col_base = blockIdx.x * 16;
  const int col  = col_base + lane16;
  const int koff = halo * 2;

  if (threadIdx.x < 16) {
    float nm = 0.999f * max_distance[col_base + threadIdx.x]
             + 0.001f * batch_max[col_base + threadIdx.x];
    s_nm[threadIdx.x] = fmaxf(nm, 1e-8f);
  }
  __syncthreads();

  const float nmax  = s_nm[lane16];
  const float ls    = log_sigma[0];
  const float alpha = -0.5f * __expf(-2.f * ls);  // -1/(2*exp(ls)^2)
  const int   p     = distance_p[0];
  const float e_lo  = 2.f / (float)p;
  const float e_hi  = 2.f * (float)p;

  v2f bfrag[16];
  float en = 0.f;
#pragma unroll
  for (int kk = 0; kk < 16; ++kk) {
    int k = kk * 4 + koff;
    float b0 = e[col * ZD + k];
    float b1 = e[col * ZD + k + 1];
    bfrag[kk].x = b0; bfrag[kk].y = b1;
    en += b0 * b0 + b1 * b1;
  }
  en += __shfl_xor(en, 16, 32);

  float mx = -INFINITY;   // running max of alpha*f(dist) for this column
  float sm = 0.f;         // running sum of exp(x - mx)

  for (int rt = wave; rt < ROW_TILES; rt += WAVES) {
    const int row_base = rt * 16;
    const int row = row_base + lane16;
    const float* zrow = z + (row >> 10) * (ZD * HW) + (row & (HW - 1));

    v8f acc = {0.f, 0.f, 0.f, 0.f, 0.f, 0.f, 0.f, 0.f};
    float zn = 0.f;
#pragma unroll
    for (int kk = 0; kk < 16; ++kk) {
      int k = kk * 4 + koff;
      float a0 = zrow[k * HW];
      float a1 = zrow[(k + 1) * HW];
      v2f a; a.x = a0; a.y = a1;
      zn += a0 * a0 + a1 * a1;
      acc = wmma_f32(a, bfrag[kk], acc);
    }
    zn += __shfl_xor(zn, 16, 32);

#pragma unroll
    for (int j = 0; j < 8; ++j) {
      int rj = j + halo * 8;
      float znj  = __shfl(zn, rj, 32);
      float d2   = znj + en - 2.f * acc[j];
      float dist = sqrtf(fmaxf(d2, 0.f));
      float f;
      if (p == 2) {                       // fast path: dist^1 / dist^4
        float ds = dist * dist;
        f = (dist < nmax) ? dist : ds * ds;
      } else {
        f = (dist < nmax) ? __powf(dist, e_lo) : __powf(dist, e_hi);
      }
      float x = alpha * f;
      if (x > mx) {                       // streaming logsumexp update
        sm = sm * __expf(mx - x) + 1.f;
        mx = x;
      } else {
        sm += __expf(x - mx);
      }
    }
  }

  // merge the two half-lane states (same column, different rows)
  {
    float mx2 = __shfl_xor(mx, 16, 32);
    float sm2 = __shfl_xor(sm, 16, 32);
    float M = fmaxf(mx, mx2);
    sm = sm * __expf(mx - M) + sm2 * __expf(mx2 - M);
    mx = M;
  }
  if (lane < 16) { smx[wave][lane16] = mx; ssm[wave][lane16] = sm; }
  __syncthreads();
  if (threadIdx.x < 16) {
    float M = smx[0][threadIdx.x];
    float S = ssm[0][threadIdx.x];
#pragma unroll
    for (int w = 1; w < WAVES; ++w) {
      float m2 = smx[w][threadIdx.x], s2 = ssm[w][threadIdx.x];
      float Mn = fmaxf(M, m2);
      S = S * __expf(M - Mn) + s2 * __expf(m2 - Mn);
      M = Mn;
    }
    lse_out[col_base + threadIdx.x] = M + __logf(S);
  }
}

// ---------------------------------------------------------------------------
// Final: loss = -mean(lse) + z_dim * log_sigma
// ---------------------------------------------------------------------------
__global__ __launch_bounds__(256) void final_loss(
    const float* __restrict__ lse, const float* __restrict__ log_sigma,
    float* __restrict__ out) {
  __shared__ float red[256];
  float s = 0.f;
  for (int i = threadIdx.x; i < M_COLS; i += 256) s += lse[i];
  red[threadIdx.x] = s;
  __syncthreads();
  for (int st = 128; st > 0; st >>= 1) {
    if (threadIdx.x < st) red[threadIdx.x] += red[threadIdx.x + st];
    __syncthreads();
  }
  if (threadIdx.x == 0)
    out[0] = -(red[0] / (float)M_COLS) + (float)ZD * log_sigma[0];
}

// ---------------------------------------------------------------------------
extern "C" void kernel_launch(void* const* d_in, const int* in_sizes, int n_in,
                              void* d_out, int out_size, void* d_ws, size_t ws_size,
                              hipStream_t stream) {
  const float* z            = (const float*)d_in[0];
  const float* e            = (const float*)d_in[1];
  const float* belong       = (const float*)d_in[2];
  const float* log_sigma    = (const float*)d_in[3];
  const float* max_distance = (const float*)d_in[4];
  const int*   distance_p   = (const int*)d_in[5];
  float* out = (float*)d_out;

  float* batch_max = (float*)d_ws;        // M floats
  float* lse       = batch_max + M_COLS;  // M floats (32 KB total scratch)

  pass1_colmax<<<M_COLS / 16, 256, 0, stream>>>(z, e, belong, batch_max);
  pass3_lse<<<M_COLS / 16, 256, 0, stream>>>(z, e, batch_max, max_distance,
                                             log_sigma, distance_p, lse);
  final_loss<<<1, 256, 0, stream>>>(lse, log_sigma, out);
}